// GATBlock_41970420418007
// MI455X (gfx1250) — compile-verified
//
#include <hip/hip_runtime.h>
#include <math.h>

#define D_FEAT 128
#define HEADS 4
#define CDIM 32
#define NEG_SLOPE 0.2f
#define LN_EPS 1e-5f

typedef __attribute__((ext_vector_type(2))) float v2f;
typedef __attribute__((ext_vector_type(8))) float v8f;

// ---------------------------------------------------------------------------
// 0. init: acc (d_out) = 0, denom = 0, m = -inf
// ---------------------------------------------------------------------------
__global__ __launch_bounds__(256) void gat_init(float* __restrict__ acc,
                                                float* __restrict__ m,
                                                float* __restrict__ denom,
                                                int N) {
  size_t total = (size_t)N * D_FEAT;
  size_t nh    = (size_t)N * HEADS;
  for (size_t i = blockIdx.x * (size_t)blockDim.x + threadIdx.x; i < total;
       i += (size_t)gridDim.x * blockDim.x) {
    acc[i] = 0.0f;
    if (i < nh) {
      m[i] = -INFINITY;
      denom[i] = 0.0f;
    }
  }
}

// ---------------------------------------------------------------------------
// 1. h = x @ W  via V_WMMA_F32_16X16X4_F32 (exact fp32 matrix path)
//    One wave computes a 16x16 tile of h; 8 waves/block cover all 128 cols
//    of a 16-row block. K loop: 128/4 = 32 WMMA ops, unrolled x8.
//    A layout (16x4 f32): lanes 0-15 hold M=lane, K={0,1}; lanes 16-31 hold
//    M=lane-16, K={2,3}.  B (4x16) mirrors the K split.  D: VGPR v holds
//    row (v + 8*(lane>=16)), col (lane&15).
// ---------------------------------------------------------------------------
__global__ __launch_bounds__(256) void gat_gemm_h(const float* __restrict__ x,
                                                  const float* __restrict__ W,
                                                  float* __restrict__ h,
                                                  int N) {
  const int lane  = threadIdx.x & 31;
  const int wid   = threadIdx.x >> 5;   // 0..7 -> 16-col tile
  const int row0  = blockIdx.x * 16;
  const int col0  = wid * 16;
  const int mrow  = lane & 15;
  const int khalf = lane >> 4;          // 0 or 1

  int arow = row0 + mrow;
  if (arow >= N) arow = N - 1;          // clamp (EXEC must stay all-1 for WMMA)

  v8f acc = {};
#pragma unroll 8
  for (int k0 = 0; k0 < D_FEAT; k0 += 4) {
    const int ka = k0 + 2 * khalf;
    // A: x[arow][ka..ka+1]  (8B-aligned vector load)
    v2f a = *(const v2f*)(x + (size_t)arow * D_FEAT + ka);
    // B: W[ka + r][col0 + mrow], r = 0,1
    const float* bp = W + (size_t)ka * D_FEAT + col0 + mrow;
    v2f b;
    b.x = bp[0];
    b.y = bp[D_FEAT];
    acc = __builtin_amdgcn_wmma_f32_16x16x4_f32(
        /*neg_a=*/false, a, /*neg_b=*/false, b,
        /*c_mod=*/(short)0, acc, /*reuse_a=*/false, /*reuse_b=*/false);
  }

  const int rbase = row0 + 8 * khalf;
  float* dp = h + (size_t)rbase * D_FEAT + col0 + mrow;
  if (row0 + 16 <= N) {
    // full tile: branch-free stores (the common path; N%16==0 always hits it)
#pragma unroll
    for (int v = 0; v < 8; ++v) dp[(size_t)v * D_FEAT] = acc[v];
  } else {
#pragma unroll
    for (int v = 0; v < 8; ++v) {
      if (rbase + v < N) dp[(size_t)v * D_FEAT] = acc[v];
    }
  }
}

// ---------------------------------------------------------------------------
// 2. a_src[n][hd] = sum_c h[n][hd*32+c]*att_src[hd*32+c]   (same for a_dst)
//    One node per block (128 threads); each wave32 = one head -> pure
//    shuffle reduction, no LDS.
// ---------------------------------------------------------------------------
__global__ __launch_bounds__(128) void gat_att_logits(const float* __restrict__ h,
                                                      const float* __restrict__ att_src,
                                                      const float* __restrict__ att_dst,
                                                      float* __restrict__ a_src,
                                                      float* __restrict__ a_dst,
                                                      int N) {
  const int n = blockIdx.x;
  if (n >= N) return;
  const int t = threadIdx.x;            // 0..127
  const float hv = h[(size_t)n * D_FEAT + t];
  float ps = hv * att_src[t];
  float pd = hv * att_dst[t];
#pragma unroll
  for (int off = 16; off > 0; off >>= 1) {
    ps += __shfl_down(ps, off, 32);
    pd += __shfl_down(pd, off, 32);
  }
  if ((t & 31) == 0) {
    const int head = t >> 5;
    a_src[(size_t)n * HEADS + head] = ps;
    a_dst[(size_t)n * HEADS + head] = pd;
  }
}

// ---------------------------------------------------------------------------
// 3. per (edge,head): e = leaky_relu(a_src[src]+a_dst[dst]); cache e;
//    atomic segment max into m[dst].
//    Edge i < E comes from edge_index; i in [E, Etot) is self-loop i-E.
// ---------------------------------------------------------------------------
__global__ __launch_bounds__(256) void gat_edge_max(const int* __restrict__ ei,
                                                    int E, int Etot,
                                                    const float* __restrict__ a_src,
                                                    const float* __restrict__ a_dst,
                                                    float* __restrict__ elog,
                                                    float* __restrict__ m) {
  const size_t tid = blockIdx.x * (size_t)blockDim.x + threadIdx.x;
  const size_t total = (size_t)Etot * HEADS;
  if (tid >= total) return;
  const int e  = (int)(tid >> 2);
  const int hd = (int)(tid & 3);
  const int src = (e < E) ? ei[e] : (e - E);
  const int dst = (e < E) ? ei[E + e] : (e - E);
  float v = a_src[(size_t)src * HEADS + hd] + a_dst[(size_t)dst * HEADS + hd];
  v = (v > 0.0f) ? v : NEG_SLOPE * v;
  elog[tid] = v;
  atomicMax(&m[(size_t)dst * HEADS + hd], v);
}

// ---------------------------------------------------------------------------
// 4. per (edge,head): ex = exp(e - m[dst]); cache ex; denom[dst] += ex
// ---------------------------------------------------------------------------
__global__ __launch_bounds__(256) void gat_edge_exp(const int* __restrict__ ei,
                                                    int E, int Etot,
                                                    float* __restrict__ elog,
                                                    const float* __restrict__ m,
                                                    float* __restrict__ denom) {
  const size_t tid = blockIdx.x * (size_t)blockDim.x + threadIdx.x;
  const size_t total = (size_t)Etot * HEADS;
  if (tid >= total) return;
  const int e  = (int)(tid >> 2);
  const int hd = (int)(tid & 3);
  const int dst = (e < E) ? ei[E + e] : (e - E);
  const float ex = __expf(elog[tid] - m[(size_t)dst * HEADS + hd]);
  elog[tid] = ex;
  atomicAdd(&denom[(size_t)dst * HEADS + hd], ex);
}

// ---------------------------------------------------------------------------
// 5. aggregation: one wave per edge. lanes = channel, loop over heads.
//    acc[dst] += h[src] * alpha   (alpha = ex / (denom+1e-16))
// ---------------------------------------------------------------------------
__global__ __launch_bounds__(256) void gat_aggregate(const int* __restrict__ ei,
                                                     int E, int Etot,
                                                     const float* __restrict__ h,
                                                     const float* __restrict__ eexp,
                                                     const float* __restrict__ denom,
                                                     float* __restrict__ acc) {
  const int lane = threadIdx.x & 31;
  const int e = blockIdx.x * (blockDim.x >> 5) + (threadIdx.x >> 5);
  if (e >= Etot) return;
  const int src = (e < E) ? ei[e] : (e - E);
  const int dst = (e < E) ? ei[E + e] : (e - E);
#pragma unroll
  for (int hd = 0; hd < HEADS; ++hd) {
    const float w = eexp[(size_t)e * HEADS + hd] /
                    (denom[(size_t)dst * HEADS + hd] + 1e-16f);
    const int c = hd * CDIM + lane;
    const float v = h[(size_t)src * D_FEAT + c] * w;
    atomicAdd(&acc[(size_t)dst * D_FEAT + c], v);
  }
}

// ---------------------------------------------------------------------------
// 6. finish: GELU(exact erf) -> LayerNorm -> + x residual, in place over acc.
//    One node per block (128 threads, 4 waves): shuffle + LDS reduction.
// ---------------------------------------------------------------------------
__global__ __launch_bounds__(128) void gat_finish(float* __restrict__ acc,
                                                  const float* __restrict__ bias,
                                                  const float* __restrict__ gamma,
                                                  const float* __restrict__ beta,
                                                  const float* __restrict__ x,
                                                  int N) {
  __shared__ float s_sum[4];
  __shared__ float s_sq[4];
  const int n = blockIdx.x;
  if (n >= N) return;
  const int t = threadIdx.x;

  const float v = acc[(size_t)n * D_FEAT + t] + bias[t];
  // exact GELU: 0.5*v*(1+erf(v/sqrt(2)))
  const float g = 0.5f * v * (1.0f + erff(v * 0.70710678118654752f));

  float s = g, sq = g * g;
#pragma unroll
  for (int off = 16; off > 0; off >>= 1) {
    s  += __shfl_down(s, off, 32);
    sq += __shfl_down(sq, off, 32);
  }
  if ((t & 31) == 0) {
    s_sum[t >> 5] = s;
    s_sq[t >> 5]  = sq;
  }
  __syncthreads();
  const float tot  = s_sum[0] + s_sum[1] + s_sum[2] + s_sum[3];
  const float totq = s_sq[0] + s_sq[1] + s_sq[2] + s_sq[3];
  const float mu  = tot * (1.0f / D_FEAT);
  const float var = totq * (1.0f / D_FEAT) - mu * mu;
  const float inv = rsqrtf(var + LN_EPS);

  const float out = (g - mu) * inv * gamma[t] + beta[t] + x[(size_t)n * D_FEAT + t];
  acc[(size_t)n * D_FEAT + t] = out;
}

// ---------------------------------------------------------------------------
// launch
// ---------------------------------------------------------------------------
extern "C" void kernel_launch(void* const* d_in, const int* in_sizes, int n_in,
                              void* d_out, int out_size, void* d_ws, size_t ws_size,
                              hipStream_t stream) {
  const float* x       = (const float*)d_in[0];
  const int*   ei      = (const int*)d_in[1];   // edge_index [2,E] (int32 after JAX canonicalization)
  const float* W       = (const float*)d_in[2];
  const float* att_src = (const float*)d_in[3];
  const float* att_dst = (const float*)d_in[4];
  const float* bias    = (const float*)d_in[5];
  const float* gamma   = (const float*)d_in[6];
  const float* beta    = (const float*)d_in[7];

  const int N    = in_sizes[0] / D_FEAT;
  const int E    = in_sizes[1] / 2;
  const int Etot = E + N;

  float* ws     = (float*)d_ws;
  float* h      = ws;                                  // N*128
  float* a_src  = h + (size_t)N * D_FEAT;              // N*4
  float* a_dst  = a_src + (size_t)N * HEADS;           // N*4
  float* mbuf   = a_dst + (size_t)N * HEADS;           // N*4
  float* denom  = mbuf + (size_t)N * HEADS;            // N*4
  float* elog   = denom + (size_t)N * HEADS;           // Etot*4 (logits -> exp)
  float* acc    = (float*)d_out;                       // N*128, in-place output

  // 0. init
  {
    size_t total = (size_t)N * D_FEAT;
    int blocks = (int)((total + 255) / 256);
    if (blocks > 65535) blocks = 65535;
    gat_init<<<blocks, 256, 0, stream>>>(acc, mbuf, denom, N);
  }
  // 1. h = x @ W (WMMA fp32)
  {
    int rowTiles = (N + 15) / 16;
    gat_gemm_h<<<rowTiles, 256, 0, stream>>>(x, W, h, N);
  }
  // 2. attention logits per node
  gat_att_logits<<<N, 128, 0, stream>>>(h, att_src, att_dst, a_src, a_dst, N);
  // 3. per-edge logit + segment max
  {
    size_t total = (size_t)Etot * HEADS;
    int blocks = (int)((total + 255) / 256);
    gat_edge_max<<<blocks, 256, 0, stream>>>(ei, E, Etot, a_src, a_dst, elog, mbuf);
  }
  // 4. exp + denom
  {
    size_t total = (size_t)Etot * HEADS;
    int blocks = (int)((total + 255) / 256);
    gat_edge_exp<<<blocks, 256, 0, stream>>>(ei, E, Etot, elog, mbuf, denom);
  }
  // 5. weighted scatter-add aggregation (one wave per edge)
  {
    int edgesPerBlock = 256 / 32;
    int blocks = (Etot + edgesPerBlock - 1) / edgesPerBlock;
    gat_aggregate<<<blocks, 256, 0, stream>>>(ei, E, Etot, h, elog, denom, acc);
  }
  // 6. GELU -> LayerNorm -> residual
  gat_finish<<<N, 128, 0, stream>>>(acc, bias, gamma, beta, x, N);
}